// GraphAttentionLayerSkip_45028437131375
// MI455X (gfx1250) — compile-verified
//
#include <hip/hip_runtime.h>
#include <hip/hip_bf16.h>
#include <math.h>

#define N_NODES 50000
#define N_EDGES 800000
#define N_FEAT  128
#define N_HID   64
#define NEG_SLOPE 0.2f
#define LN_EPS  1e-5f

typedef __attribute__((ext_vector_type(2))) float v2f;
typedef __attribute__((ext_vector_type(8))) float v8f;

__device__ __forceinline__ float wave_sum32(float v) {
#pragma unroll
    for (int off = 16; off > 0; off >>= 1)
        v += __shfl_xor(v, off, 32);
    return v;
}

// order-preserving float -> uint encoding for atomic max
__device__ __forceinline__ unsigned enc_f32(float x) {
    unsigned u = __float_as_uint(x);
    return (u & 0x80000000u) ? ~u : (u | 0x80000000u);
}
__device__ __forceinline__ float dec_f32(unsigned e) {
    unsigned u = (e & 0x80000000u) ? (e ^ 0x80000000u) : ~e;
    return __uint_as_float(u);
}

// ---------------------------------------------------------------------------
// Kernel 1: h = X @ W^T + bias via V_WMMA_F32_16X16X4_F32.
// One wave computes a 16x64 output tile (4 N-tiles of 16x16, K=128 in 32 steps).
// Also seeds accum = h and initializes seg_max / seg_sum for this tile's rows.
// ---------------------------------------------------------------------------
__global__ __launch_bounds__(256) void gat_wmma_gemm(
    const float* __restrict__ X, const float* __restrict__ W,
    const float* __restrict__ bias,
    float* __restrict__ h, float* __restrict__ accum,
    unsigned* __restrict__ seg_max, float* __restrict__ seg_sum)
{
    __shared__ float lw[N_HID * N_FEAT]; // 32 KB: whole W in LDS
    for (int i = threadIdx.x; i < N_HID * N_FEAT; i += 256)
        lw[i] = W[i];
    __syncthreads();

    const int lane = threadIdx.x & 31;
    const int wave = blockIdx.x * 8 + (threadIdx.x >> 5);
    const int m0   = wave * 16;
    if (m0 >= N_NODES) return;

    const int sub16 = lane & 15;           // M for A, N for B/C/D
    const int kSub  = (lane >> 4) * 2;     // {0,2}: K pair held by this lane half
    const int mHi   = (lane >> 4) * 8;     // C/D row offset for upper lanes

    v8f acc[4] = {};                       // 4 N-tiles of 16x16 f32

    const float* xrow = X + (size_t)(m0 + sub16) * N_FEAT;

    for (int k0 = 0; k0 < N_FEAT; k0 += 4) {
        v2f a;
        a.x = xrow[k0 + kSub];
        a.y = xrow[k0 + kSub + 1];
#pragma unroll
        for (int t = 0; t < 4; ++t) {
            const float* wrow = lw + (size_t)(t * 16 + sub16) * N_FEAT;
            v2f b;
            b.x = wrow[k0 + kSub];
            b.y = wrow[k0 + kSub + 1];
            acc[t] = __builtin_amdgcn_wmma_f32_16x16x4_f32(
                false, a, false, b, (short)0, acc[t], false, false);
        }
    }

#pragma unroll
    for (int t = 0; t < 4; ++t) {
        const int n  = t * 16 + sub16;
        const float bn = bias[n];
#pragma unroll
        for (int r = 0; r < 8; ++r) {
            const int m = m0 + mHi + r;
            const float v = acc[t][r] + bn;
            h[(size_t)m * N_HID + n]     = v;
            accum[(size_t)m * N_HID + n] = v;   // skip connection seed
        }
    }

    if (lane < 16) {                       // init softmax state for these rows
        seg_max[m0 + lane] = 0u;           // < enc(-inf): safe identity
        seg_sum[m0 + lane] = 0.f;
    }
}

// ---------------------------------------------------------------------------
// Kernel 2: per-node attention scalars p = h.a1, q = h.a2 (wave per node)
// ---------------------------------------------------------------------------
__global__ __launch_bounds__(256) void gat_node_pq(
    const float* __restrict__ h, const float* __restrict__ attn_w,
    float* __restrict__ p, float* __restrict__ q)
{
    const int lane = threadIdx.x & 31;
    const int node = blockIdx.x * 8 + (threadIdx.x >> 5);
    if (node >= N_NODES) return;
    const float h0 = h[(size_t)node * N_HID + lane];
    const float h1 = h[(size_t)node * N_HID + lane + 32];
    float ps = h0 * attn_w[lane] + h1 * attn_w[lane + 32];
    float qs = h0 * attn_w[N_HID + lane] + h1 * attn_w[N_HID + lane + 32];
    ps = wave_sum32(ps);
    qs = wave_sum32(qs);
    if (lane == 0) { p[node] = ps; q[node] = qs; }
}

// ---------------------------------------------------------------------------
// Kernel 3: edge logits + LeakyReLU + segment max (hardware u32 atomic max)
// ---------------------------------------------------------------------------
__global__ __launch_bounds__(256) void gat_edge_score(
    const int* __restrict__ tgt, const int* __restrict__ src,
    const float* __restrict__ p, const float* __restrict__ q,
    const float* __restrict__ attn_b,
    float* __restrict__ ebuf, unsigned* __restrict__ seg_max)
{
    const int i = blockIdx.x * blockDim.x + threadIdx.x;
    if (i >= N_EDGES) return;
    const int t = tgt[i], s = src[i];
    float e = p[t] + q[s] + attn_b[0];
    e = (e >= 0.f) ? e : NEG_SLOPE * e;
    ebuf[i] = e;
    atomicMax(seg_max + t, enc_f32(e));
}

// ---------------------------------------------------------------------------
// Kernel 4: ex = exp(e - segmax[tgt]); segment sum (hardware f32 atomic add)
// ---------------------------------------------------------------------------
__global__ __launch_bounds__(256) void gat_edge_exp(
    const int* __restrict__ tgt,
    const unsigned* __restrict__ seg_max, float* __restrict__ ebuf,
    float* __restrict__ seg_sum)
{
    const int i = blockIdx.x * blockDim.x + threadIdx.x;
    if (i >= N_EDGES) return;
    const int t = tgt[i];
    const float ex = expf(ebuf[i] - dec_f32(seg_max[t]));
    ebuf[i] = ex;
    __hip_atomic_fetch_add(seg_sum + t, ex, __ATOMIC_RELAXED,
                           __HIP_MEMORY_SCOPE_AGENT);
}

// ---------------------------------------------------------------------------
// Kernel 5: message scatter, wave per edge: accum[tgt] += alpha * h[src]
// (64 features: 2 per lane, coalesced 128B atomic-f32 bursts)
// ---------------------------------------------------------------------------
__global__ __launch_bounds__(256) void gat_edge_scatter(
    const int* __restrict__ tgt, const int* __restrict__ src,
    const float* __restrict__ ebuf, const float* __restrict__ seg_sum,
    const float* __restrict__ h, float* __restrict__ accum)
{
    const int lane = threadIdx.x & 31;
    const int edge = blockIdx.x * 8 + (threadIdx.x >> 5);
    if (edge >= N_EDGES) return;
    const int t = tgt[edge], s = src[edge];
    const float alpha = ebuf[edge] / seg_sum[t];
    const float* hs = h + (size_t)s * N_HID;
    float* at = accum + (size_t)t * N_HID;
    __hip_atomic_fetch_add(at + lane,      alpha * hs[lane],
                           __ATOMIC_RELAXED, __HIP_MEMORY_SCOPE_AGENT);
    __hip_atomic_fetch_add(at + lane + 32, alpha * hs[lane + 32],
                           __ATOMIC_RELAXED, __HIP_MEMORY_SCOPE_AGENT);
}

// ---------------------------------------------------------------------------
// Kernel 6: ELU-ish (expm1) + LayerNorm over 64 feats (wave per node)
// ---------------------------------------------------------------------------
__global__ __launch_bounds__(256) void gat_finalize(
    const float* __restrict__ accum, const float* __restrict__ gamma,
    const float* __restrict__ beta, float* __restrict__ out)
{
    const int lane = threadIdx.x & 31;
    const int node = blockIdx.x * 8 + (threadIdx.x >> 5);
    if (node >= N_NODES) return;
    float x0 = accum[(size_t)node * N_HID + lane];
    float x1 = accum[(size_t)node * N_HID + lane + 32];
    x0 = (x0 > 0.f) ? x0 : expm1f(x0);
    x1 = (x1 > 0.f) ? x1 : expm1f(x1);
    const float mu = wave_sum32(x0 + x1) * (1.f / N_HID);
    const float d0 = x0 - mu, d1 = x1 - mu;
    const float var = wave_sum32(d0 * d0 + d1 * d1) * (1.f / N_HID);
    const float r = rsqrtf(var + LN_EPS);
    out[(size_t)node * N_HID + lane]      = d0 * r * gamma[lane]      + beta[lane];
    out[(size_t)node * N_HID + lane + 32] = d1 * r * gamma[lane + 32] + beta[lane + 32];
}

extern "C" void kernel_launch(void* const* d_in, const int* in_sizes, int n_in,
                              void* d_out, int out_size, void* d_ws, size_t ws_size,
                              hipStream_t stream) {
    const float* X     = (const float*)d_in[0];  // [50000,128]
    const float* W     = (const float*)d_in[1];  // [64,128]
    const float* Wb    = (const float*)d_in[2];  // [64]
    const float* Aw    = (const float*)d_in[3];  // [1,128]
    const float* Ab    = (const float*)d_in[4];  // [1]
    const float* gamma = (const float*)d_in[5];  // [64]
    const float* beta  = (const float*)d_in[6];  // [64]
    const int*   ei    = (const int*)d_in[7];    // [2,800000]
    const int* tgt = ei;
    const int* src = ei + N_EDGES;
    float* out = (float*)d_out;

    char* ws = (char*)d_ws;
    float*    h     = (float*)ws;    ws += (size_t)N_NODES * N_HID * sizeof(float);
    float*    accum = (float*)ws;    ws += (size_t)N_NODES * N_HID * sizeof(float);
    float*    p     = (float*)ws;    ws += (size_t)N_NODES * sizeof(float);
    float*    q     = (float*)ws;    ws += (size_t)N_NODES * sizeof(float);
    unsigned* smax  = (unsigned*)ws; ws += (size_t)N_NODES * sizeof(unsigned);
    float*    ssum  = (float*)ws;    ws += (size_t)N_NODES * sizeof(float);
    float*    ebuf  = (float*)ws;    ws += (size_t)N_EDGES * sizeof(float);
    (void)ws_size; (void)in_sizes; (void)n_in; (void)out_size;

    const int tiles = (N_NODES + 15) / 16;                 // 3125
    gat_wmma_gemm<<<(tiles + 7) / 8, 256, 0, stream>>>(X, W, Wb, h, accum, smax, ssum);
    gat_node_pq<<<(N_NODES + 7) / 8, 256, 0, stream>>>(h, Aw, p, q);
    gat_edge_score<<<(N_EDGES + 255) / 256, 256, 0, stream>>>(tgt, src, p, q, Ab, ebuf, smax);
    gat_edge_exp<<<(N_EDGES + 255) / 256, 256, 0, stream>>>(tgt, smax, ebuf, ssum);
    gat_edge_scatter<<<(N_EDGES + 7) / 8, 256, 0, stream>>>(tgt, src, ebuf, ssum, h, accum);
    gat_finalize<<<(N_NODES + 7) / 8, 256, 0, stream>>>(accum, gamma, beta, out);
}